// CLNModel_87686052315398
// MI455X (gfx1250) — compile-verified
//
#include <hip/hip_runtime.h>
#include <stdint.h>
#include <stddef.h>

#define CLN_B      100.0f
#define CLN_EPS    0.01f
#define CLN_LOG2E  1.4426950408889634f
#define ROWS_PER_BLOCK 32
#define NWAVES 8                       // waves per block, each owns a clause range
#define BLOCK_THREADS (ROWS_PER_BLOCK * NWAVES)

// ---- CDNA5 async global->LDS path (probe via __has_builtin; safe fallback) ----
#if defined(__HIP_DEVICE_COMPILE__)
#  if __has_builtin(__builtin_amdgcn_global_load_async_to_lds_b32)
#    define CLN_ASYNC_LDS 1
typedef __attribute__((address_space(1))) int cln_as1_int;
typedef __attribute__((address_space(3))) int cln_as3_int;
#  endif
#endif

__device__ __forceinline__ float cln_exp2(float x) {
#if defined(__HIP_DEVICE_COMPILE__) && __has_builtin(__builtin_amdgcn_exp2f)
    return __builtin_amdgcn_exp2f(x);      // v_exp_f32 (base 2)
#else
    return exp2f(x);
#endif
}

__device__ __forceinline__ float cln_log2(float x) {
#if defined(__HIP_DEVICE_COMPILE__) && __has_builtin(__builtin_amdgcn_logf)
    return __builtin_amdgcn_logf(x);       // v_log_f32 (base 2)
#else
    return log2f(x);
#endif
}

// Scalar (wave-uniform) lower_bound on the sorted clause_ids array.
__device__ __forceinline__ int cln_lower_bound(const int* __restrict__ a,
                                               int n, int target) {
    int lo = 0, hi = n;
    while (lo < hi) {
        int mid = (lo + hi) >> 1;
        if (a[mid] < target) lo = mid + 1; else hi = mid;
    }
    return lo;
}

// Block = 8 waves x 32 lanes. Lane = batch row (32 rows staged once in LDS,
// shared by all 8 waves). Each wave owns clauses [c0,c1) and the matching atom
// sub-range found by two scalar binary searches; atom parameters are indexed by
// the uniform loop counter => s_load + scalar branches, zero divergence.
// x-tile row stride = nfeat+5 floats => bank(lane,f) = (5*lane+f)%64,
// gcd(5,64)=1 => conflict-free ds_load gathers in the hot loop.
extern "C" __global__ void __launch_bounds__(BLOCK_THREADS)
cln_forward(const float* __restrict__ x,
            const float* __restrict__ w,
            const float* __restrict__ eta,
            const float* __restrict__ leaf,
            const float* __restrict__ gate,
            const int*   __restrict__ feat_idx,
            const int*   __restrict__ cmp_sign,
            const int*   __restrict__ clause_ids,
            float*       __restrict__ out,
            int batch, int nfeat, int n_atoms, int n_clauses)
{
    extern __shared__ float smem[];
    float* tile = smem;                                    // 32 x (nfeat+5)
    const int stride = nfeat + 5;
    float* ypart = smem + ROWS_PER_BLOCK * stride;         // NWAVES x 32

    const int tid     = threadIdx.x;
    const int lane    = tid & (ROWS_PER_BLOCK - 1);
    const int wave_id = tid >> 5;
    const int row0    = blockIdx.x * ROWS_PER_BLOCK;

    // ---- Stage 32 rows of x into LDS (async DMA path on CDNA5) ----
    for (int r = 0; r < ROWS_PER_BLOCK; ++r) {
        const int gr = row0 + r;
        if (gr >= batch) break;
        const float* src = x + (size_t)gr * nfeat;
        float*       dst = tile + r * stride;
        for (int f = tid; f < nfeat; f += BLOCK_THREADS) {
#if defined(CLN_ASYNC_LDS)
            __builtin_amdgcn_global_load_async_to_lds_b32(
                (cln_as1_int*)(src + f),
                (cln_as3_int*)(dst + f),
                /*offset=*/0, /*cpol=*/0);
#else
            dst[f] = src[f];
#endif
        }
    }
#if defined(CLN_ASYNC_LDS)
#  if __has_builtin(__builtin_amdgcn_s_wait_asynccnt)
    __builtin_amdgcn_s_wait_asynccnt(0);
#  else
    asm volatile("s_wait_asynccnt 0" ::: "memory");
#  endif
#endif
    __syncthreads();

    // ---- This wave's clause range and matching atom sub-range (all scalar) ----
    const int c0 = (int)(((long long)n_clauses * wave_id)       / NWAVES);
    const int c1 = (int)(((long long)n_clauses * (wave_id + 1)) / NWAVES);
    const int a0 = cln_lower_bound(clause_ids, n_atoms, c0);
    const int a1 = cln_lower_bound(clause_ids, n_atoms, c1);

    const float* xr = tile + lane * stride;

    // log(sigmoid(z)) = -ln2*log2(1 + 2^{-z*log2e}); with z = sign*B*(t+eps):
    //   arg = m*(w*x - eta + eps) = (m*w)*x + m*(eps-eta),
    //   m = +B*log2e for '<' (cmp==0), -B*log2e for '>=' (cmp==1).
    // acc accumulates log2(1+2^{arg}) >= 0; clause value = 2^{-acc}.
    float y   = 0.0f;
    float acc = 0.0f;
    int   cur = c0 - 1;

    for (int a = a0; a < a1; ++a) {
        const int cid = clause_ids[a];                  // uniform -> SGPR
        if (cid != cur) {                               // scalar branch
            if (cur >= c0)
                y = fmaf(gate[cur] * leaf[cur], cln_exp2(-acc), y);
            for (int c = cur + 1; c < cid; ++c)         // empty clauses: exp(0)=1
                y += gate[c] * leaf[c];
            cur = cid;
            acc = 0.0f;
        }
        const float m  = (cmp_sign[a] == 0) ? (CLN_B * CLN_LOG2E)
                                            : (-CLN_B * CLN_LOG2E);
        const float A  = m * w[a];                      // wave-uniform (SALU)
        const float Bc = m * (CLN_EPS - eta[a]);        // wave-uniform (SALU)
        const float xv = xr[feat_idx[a]];               // conflict-free ds_load
        acc += cln_log2(1.0f + cln_exp2(fmaf(A, xv, Bc)));
    }
    if (cur >= c0)
        y = fmaf(gate[cur] * leaf[cur], cln_exp2(-acc), y);
    for (int c = cur + 1; c < c1; ++c)                  // trailing empty clauses
        y += gate[c] * leaf[c];

    // ---- Combine the 8 per-wave partials per row ----
    ypart[wave_id * ROWS_PER_BLOCK + lane] = y;
    __syncthreads();

    if (wave_id == 0) {
        float t = 0.0f;
        #pragma unroll
        for (int v = 0; v < NWAVES; ++v)
            t += ypart[v * ROWS_PER_BLOCK + lane];
        const int gr = row0 + lane;
        if (gr < batch) out[gr] = t;
    }
}

extern "C" void kernel_launch(void* const* d_in, const int* in_sizes, int n_in,
                              void* d_out, int out_size, void* d_ws, size_t ws_size,
                              hipStream_t stream)
{
    const float* x    = (const float*)d_in[0];
    const float* w    = (const float*)d_in[1];
    const float* eta  = (const float*)d_in[2];
    const float* leaf = (const float*)d_in[3];
    const float* gate = (const float*)d_in[4];
    const int*   fid  = (const int*)d_in[5];
    const int*   cmp  = (const int*)d_in[6];
    const int*   cls  = (const int*)d_in[7];
    float*       out  = (float*)d_out;

    const int batch     = out_size;                  // 8192
    const int nfeat     = in_sizes[0] / batch;       // 256
    const int n_atoms   = in_sizes[1];               // 4096
    const int n_clauses = in_sizes[3];               // 512

    const int    grid  = (batch + ROWS_PER_BLOCK - 1) / ROWS_PER_BLOCK;
    const size_t shmem = ((size_t)ROWS_PER_BLOCK * (size_t)(nfeat + 5)
                          + (size_t)ROWS_PER_BLOCK * NWAVES) * sizeof(float);

    cln_forward<<<dim3(grid), dim3(BLOCK_THREADS), shmem, stream>>>(
        x, w, eta, leaf, gate, fid, cmp, cls, out,
        batch, nfeat, n_atoms, n_clauses);
}